// Encoder_23871428231490
// MI455X (gfx1250) — compile-verified
//
#include <hip/hip_runtime.h>
#include <stdint.h>

typedef __attribute__((ext_vector_type(2))) float v2f;
typedef __attribute__((ext_vector_type(8))) float v8f;

#define F_IN 512
#define F_H  128
#define F_OUT 64

// ---------------- degree / normalization ----------------

__global__ void k_deg_init(float* __restrict__ deg, int n) {
  int i = blockIdx.x * blockDim.x + threadIdx.x;
  if (i < n) deg[i] = 1.0f;  // self-loop contribution
}

__global__ void k_deg_edges(const long long* __restrict__ dst, float* __restrict__ deg,
                            long long E) {
  long long i = (long long)blockIdx.x * blockDim.x + threadIdx.x;
  if (i < E) atomicAdd(&deg[dst[i]], 1.0f);
}

__global__ void k_dinv(float* __restrict__ deg, int n) {
  int i = blockIdx.x * blockDim.x + threadIdx.x;
  if (i < n) deg[i] = rsqrtf(deg[i]);  // deg >= 1 always (self loops)
}

// ---------------- fp32 WMMA GEMM: OUT[nrows,NCOLS] = X[nrows,K] @ W[K,NCOLS] -------------
// One 16x16 output tile per wave via V_WMMA_F32_16X16X4_F32.
// A tile staged in LDS with +4 float padding (stride 516/132 -> conflict-free:
// lane m hits bank (m*stride)%64, all distinct).

template <int K, int NCOLS>
__global__ __launch_bounds__(256) void k_gemm_wmma(const float* __restrict__ X,
                                                   const float* __restrict__ W,
                                                   float* __restrict__ OUT, int nrows) {
  constexpr int TILES_N = NCOLS / 16;      // 8 (layer1) or 4 (layer2)
  constexpr int MT_PER_BLK = 8 / TILES_N;  // 1 or 2
  constexpr int ROWS = MT_PER_BLK * 16;    // 16 or 32
  constexpr int LDK = K + 4;               // padded LDS row stride
  __shared__ float ldsX[ROWS * LDK];

  const int tid = threadIdx.x;
  const int wave = tid >> 5;
  const int lane = tid & 31;
  const int row0 = blockIdx.x * ROWS;
  if (row0 >= nrows) return;
  const bool full = (row0 + ROWS) <= nrows;  // block-uniform -> scalar branch

  // stage A tile: coalesced global reads (consecutive tid -> consecutive k)
  if (full) {
    for (int idx = tid; idx < ROWS * K; idx += 256) {
      int r = idx / K;
      int k = idx - r * K;
      ldsX[r * LDK + k] = X[(size_t)(row0 + r) * K + k];
    }
  } else {
    for (int idx = tid; idx < ROWS * K; idx += 256) {
      int r = idx / K;
      int k = idx - r * K;
      int gr = row0 + r;
      ldsX[r * LDK + k] = (gr < nrows) ? X[(size_t)gr * K + k] : 0.0f;
    }
  }
  __syncthreads();

  const int mtile = wave / TILES_N;
  const int ntile = wave % TILES_N;
  const int lm = lane & 15;           // M (for A) / N (for B) within tile
  const int khalf = (lane >> 4) << 1; // lanes 0-15 -> K+0/+1 ; lanes 16-31 -> K+2/+3

  const float* arow = &ldsX[(mtile * 16 + lm) * LDK];
  const float* wcol = W + ntile * 16 + lm;

  v8f c = {};
  for (int k0 = 0; k0 < K; k0 += 4) {
    v2f a, b;
    a.x = arow[k0 + khalf];                        // ds_load_2addr_b64 (8B aligned)
    a.y = arow[k0 + khalf + 1];
    b.x = wcol[(size_t)(k0 + khalf) * NCOLS];      // L2-resident weight reads
    b.y = wcol[(size_t)(k0 + khalf + 1) * NCOLS];
    // D = A(16x4) x B(4x16) + C
    c = __builtin_amdgcn_wmma_f32_16x16x4_f32(false, a, false, b, (short)0, c,
                                              false, false);
  }

  // D layout: vgpr v holds M = v + 8*(lane>=16), N = lane&15
  const int mbase = mtile * 16 + ((lane >> 4) << 3);
  float* outp = OUT + (size_t)(row0 + mbase) * NCOLS + ntile * 16 + lm;
  if (full) {
#pragma unroll
    for (int v = 0; v < 8; ++v) outp[(size_t)v * NCOLS] = c[v];
  } else {
#pragma unroll
    for (int v = 0; v < 8; ++v) {
      if (row0 + mbase + v < nrows) outp[(size_t)v * NCOLS] = c[v];
    }
  }
}

// ---------------- aggregation ----------------

// hout[i] = hpre[i] * dinv[node]^2   (self-loop term; also fully initializes hout)
template <int F>
__global__ void k_selfloop_init(const float* __restrict__ hpre, const float* __restrict__ dinv,
                                float* __restrict__ hout, int n) {
  long long i = (long long)blockIdx.x * blockDim.x + threadIdx.x;
  if (i < (long long)n * F) {
    int node = (int)(i / F);  // F is a power of two -> shift
    float d = dinv[node];
    hout[i] = hpre[i] * d * d;
  }
}

// one wave per edge; lane handles F/32 consecutive floats; atomics land in L2 (hout fits)
template <int F>
__global__ __launch_bounds__(256) void k_agg_edges(const long long* __restrict__ src,
                                                   const long long* __restrict__ dst,
                                                   const float* __restrict__ dinv,
                                                   const float* __restrict__ hpre,
                                                   float* __restrict__ hout, long long E) {
  long long w = ((long long)blockIdx.x * blockDim.x + threadIdx.x) >> 5;
  int lane = threadIdx.x & 31;
  if (w >= E) return;
  long long s = src[w];
  long long d = dst[w];
  float nw = dinv[s] * dinv[d];
  constexpr int PER = F / 32;  // 4 (layer1) or 2 (layer2)
  const float* hp = hpre + (size_t)s * F + lane * PER;
  float* op = hout + (size_t)d * F + lane * PER;
  if constexpr (PER == 4) {
    float4 v = *(const float4*)hp;
    atomicAdd(op + 0, v.x * nw);
    atomicAdd(op + 1, v.y * nw);
    atomicAdd(op + 2, v.z * nw);
    atomicAdd(op + 3, v.w * nw);
  } else {
    float2 v = *(const float2*)hp;
    atomicAdd(op + 0, v.x * nw);
    atomicAdd(op + 1, v.y * nw);
  }
}

template <int F>
__global__ void k_bias_relu(float* __restrict__ h, const float* __restrict__ b, int n) {
  long long i = (long long)blockIdx.x * blockDim.x + threadIdx.x;
  if (i < (long long)n * F) {
    float v = h[i] + b[i & (F - 1)];
    h[i] = v > 0.0f ? v : 0.0f;
  }
}

// ---------------- launch ----------------

extern "C" void kernel_launch(void* const* d_in, const int* in_sizes, int n_in,
                              void* d_out, int out_size, void* d_ws, size_t ws_size,
                              hipStream_t stream) {
  const float* x      = (const float*)d_in[0];
  const long long* ei = (const long long*)d_in[1];  // int64 edge_index [2, E]
  const float* W1     = (const float*)d_in[2];
  const float* b1     = (const float*)d_in[3];
  const float* W2     = (const float*)d_in[4];
  const float* b2     = (const float*)d_in[5];

  const int N = in_sizes[0] / F_IN;
  const long long E = (long long)in_sizes[1] / 2;
  const long long* src = ei;      // edge_index[0]
  const long long* dstp = ei + E; // edge_index[1]

  // workspace layout: dinv[N] | hpre/zpre[N*128] | h1[N*128]
  char* ws = (char*)d_ws;
  float* dinv = (float*)ws;
  size_t off = (((size_t)N * 4) + 255) & ~(size_t)255;
  float* hpre = (float*)(ws + off);             // reused as zpre for layer 2
  size_t off2 = off + (size_t)N * F_H * 4;
  float* h1 = (float*)(ws + off2);
  float* out = (float*)d_out;

  const int T = 256;
  unsigned gN   = (unsigned)((N + T - 1) / T);
  unsigned gE   = (unsigned)((E + T - 1) / T);
  unsigned gEw  = (unsigned)((E * 32 + T - 1) / T);  // one wave per edge
  unsigned gNH  = (unsigned)(((long long)N * F_H + T - 1) / T);
  unsigned gNO  = (unsigned)(((long long)N * F_OUT + T - 1) / T);

  // normalization
  k_deg_init<<<gN, T, 0, stream>>>(dinv, N);
  k_deg_edges<<<gE, T, 0, stream>>>(dstp, dinv, E);
  k_dinv<<<gN, T, 0, stream>>>(dinv, N);

  // layer 1
  k_gemm_wmma<F_IN, F_H><<<(unsigned)((N + 15) / 16), 256, 0, stream>>>(x, W1, hpre, N);
  k_selfloop_init<F_H><<<gNH, T, 0, stream>>>(hpre, dinv, h1, N);
  k_agg_edges<F_H><<<gEw, T, 0, stream>>>(src, dstp, dinv, hpre, h1, E);
  k_bias_relu<F_H><<<gNH, T, 0, stream>>>(h1, b1, N);

  // layer 2 (zpre reuses hpre buffer)
  k_gemm_wmma<F_H, F_OUT><<<(unsigned)((N + 31) / 32), 256, 0, stream>>>(h1, W2, hpre, N);
  k_selfloop_init<F_OUT><<<gNO, T, 0, stream>>>(hpre, dinv, out, N);
  k_agg_edges<F_OUT><<<gEw, T, 0, stream>>>(src, dstp, dinv, hpre, out, E);
  k_bias_relu<F_OUT><<<gNO, T, 0, stream>>>(out, b2, N);
}